// MoEFusedTKG_53025666236534
// MI455X (gfx1250) — compile-verified
//
#include <hip/hip_runtime.h>

#define H_DIM 2048
#define E_NUM 16
#define TOPK 2
#define I_DIM 1408
#define IS_DIM 5632
#define T_NUM 4

// Reduction-split factors (waves per block, each owning a K-chunk)
#define WAVES_RG 8    // routed gate/up:  H split 8 ways  (chunk 256)
#define WAVES_RD 8    // routed down:     k(2) x I/4 ways (chunk 352)
#define WAVES_SG 16   // shared gate/up:  H split 16 ways (chunk 128)
#define WAVES_SD 16   // shared down:     IS split 16 ways (chunk 352)

typedef __attribute__((ext_vector_type(2))) float v2f;
typedef __attribute__((ext_vector_type(8))) float v8f;

__device__ __forceinline__ v8f wmma_f32_16x16x4(v2f a, v2f b, v8f c) {
  // D = A(16x4 f32) * B(4x16 f32) + C(16x16 f32)
  return __builtin_amdgcn_wmma_f32_16x16x4_f32(
      /*neg_a=*/false, a, /*neg_b=*/false, b,
      /*c_mod=*/(short)0, c, /*reuse_a=*/false, /*reuse_b=*/false);
}

__device__ __forceinline__ float silu_f(float v) {
  return v / (1.0f + __expf(-v));
}

// ---------------------------------------------------------------------------
// Kernel 1: router logits -> softmax -> top-2 (tiny; 1 block)
// ---------------------------------------------------------------------------
__global__ void router_topk_kernel(const float* __restrict__ x,
                                   const float* __restrict__ rw,
                                   int* __restrict__ idx_out,
                                   float* __restrict__ aff_out) {
  __shared__ float logits[T_NUM * E_NUM];
  const int tid = threadIdx.x;  // blockDim = 64
  if (tid < T_NUM * E_NUM) {
    const int t = tid / E_NUM;
    const int e = tid % E_NUM;
    const float* xt = x + t * H_DIM;
    const float* we = rw + e * H_DIM;
    float acc = 0.0f;
    for (int h = 0; h < H_DIM; ++h) acc += xt[h] * we[h];
    logits[tid] = acc;
  }
  __syncthreads();
  if (tid < T_NUM) {
    float m = -1.0e30f;
    for (int e = 0; e < E_NUM; ++e) m = fmaxf(m, logits[tid * E_NUM + e]);
    float aff[E_NUM];
    float s = 0.0f;
    for (int e = 0; e < E_NUM; ++e) {
      aff[e] = __expf(logits[tid * E_NUM + e] - m);
      s += aff[e];
    }
    const float inv = 1.0f / s;
    float b0 = -1.0f, b1 = -1.0f;
    int i0 = 0, i1 = 0;
    for (int e = 0; e < E_NUM; ++e) {
      const float v = aff[e] * inv;
      if (v > b0) { b1 = b0; i1 = i0; b0 = v; i0 = e; }
      else if (v > b1) { b1 = v; i1 = e; }
    }
    idx_out[tid * TOPK + 0] = i0;
    idx_out[tid * TOPK + 1] = i1;
    aff_out[tid * TOPK + 0] = b0;
    aff_out[tid * TOPK + 1] = b1;
  }
}

// ---------------------------------------------------------------------------
// Kernel 2: routed gate/up GEMV + SiLU*up, affinity folded in.
// Weights in B (lane = contiguous i), broadcast token in A.
// grid (I/16, T*K), block = 8 waves; wave w reduces h in [w*256, w*256+256).
// ---------------------------------------------------------------------------
__global__ void routed_gateup_kernel(const float* __restrict__ x,
                                     const float* __restrict__ gup,
                                     const int* __restrict__ idx,
                                     const float* __restrict__ aff,
                                     float* __restrict__ ws_h) {
  __shared__ float red_g[WAVES_RG * 16];
  __shared__ float red_u[WAVES_RG * 16];
  const int lane = threadIdx.x & 31;
  const int wv = threadIdx.x >> 5;
  const int n = lane & 15;              // B column / output feature
  const int bk = (lane >> 4) * 2;       // K pair selector per ISA layout
  const int tk = blockIdx.y;
  const int t = tk >> 1;
  const int e = idx[tk];
  const float a_scale = aff[tk];
  const int ib = blockIdx.x * 16;

  const float* xt = x + (size_t)t * H_DIM;
  const float* wg = gup + (size_t)e * H_DIM * 2 * I_DIM;  // [h][2][i]

  const int h_beg = wv * (H_DIM / WAVES_RG);
  const int h_end = h_beg + (H_DIM / WAVES_RG);

  v8f dg = {};
  v8f du = {};
  for (int h0 = h_beg; h0 < h_end; h0 += 4) {
    // A: all 16 rows identical = x[t][h0+bk .. h0+bk+1] (8B broadcast load)
    v2f a = *(const v2f*)(xt + h0 + bk);
    const float* wrow0 = wg + (size_t)(h0 + bk) * (2 * I_DIM);
    const float* wrow1 = wrow0 + 2 * I_DIM;
    v2f bg, bu;
    bg.x = wrow0[ib + n];            // gate row h0+bk   (coalesced over lanes)
    bg.y = wrow1[ib + n];            // gate row h0+bk+1
    bu.x = wrow0[I_DIM + ib + n];    // up   row h0+bk
    bu.y = wrow1[I_DIM + ib + n];    // up   row h0+bk+1
    dg = wmma_f32_16x16x4(a, bg, dg);
    du = wmma_f32_16x16x4(a, bu, du);
  }
  if (lane < 16) {
    red_g[wv * 16 + lane] = dg[0];
    red_u[wv * 16 + lane] = du[0];
  }
  __syncthreads();
  if (wv == 0 && lane < 16) {
    float g = 0.0f, u = 0.0f;
#pragma unroll
    for (int w = 0; w < WAVES_RG; ++w) {   // fixed order -> deterministic
      g += red_g[w * 16 + lane];
      u += red_u[w * 16 + lane];
    }
    ws_h[(size_t)tk * I_DIM + ib + lane] = a_scale * silu_f(g) * u;
  }
}

// ---------------------------------------------------------------------------
// Kernel 3: routed down, both top-k experts summed (affinity pre-folded).
// grid (H/16, T), block = 8 waves; wave w: k = w&1, i-chunk (w>>1) of 4.
// ---------------------------------------------------------------------------
__global__ void routed_down_kernel(const float* __restrict__ dwn,
                                   const int* __restrict__ idx,
                                   const float* __restrict__ ws_h,
                                   float* __restrict__ ws_routed) {
  __shared__ float red_d[WAVES_RD * 16];
  const int lane = threadIdx.x & 31;
  const int wv = threadIdx.x >> 5;
  const int n = lane & 15;
  const int bk = (lane >> 4) * 2;
  const int t = blockIdx.y;
  const int hb = blockIdx.x * 16;

  const int k = wv & 1;
  const int q = wv >> 1;                       // 0..3
  const int i_beg = q * (I_DIM / 4);           // 352-wide chunks
  const int i_end = i_beg + (I_DIM / 4);

  const int tk = t * TOPK + k;
  const int e = idx[tk];
  const float* hv = ws_h + (size_t)tk * I_DIM;
  const float* wd = dwn + (size_t)e * I_DIM * H_DIM;  // [i][h]

  v8f d = {};
  for (int i0 = i_beg; i0 < i_end; i0 += 4) {
    v2f a = *(const v2f*)(hv + i0 + bk);       // broadcast activation pair
    v2f b;
    b.x = wd[(size_t)(i0 + bk) * H_DIM + hb + n];      // coalesced rows
    b.y = wd[(size_t)(i0 + bk + 1) * H_DIM + hb + n];
    d = wmma_f32_16x16x4(a, b, d);
  }
  if (lane < 16) red_d[wv * 16 + lane] = d[0];
  __syncthreads();
  if (wv == 0 && lane < 16) {
    float acc = 0.0f;
#pragma unroll
    for (int w = 0; w < WAVES_RD; ++w) acc += red_d[w * 16 + lane];
    ws_routed[(size_t)t * H_DIM + hb + lane] = acc;
  }
}

// ---------------------------------------------------------------------------
// Kernel 4: shared gate/up. shared_{gate,up}_w is [IS, H] (h contig) ->
// weights in A (per-lane v2f row walk), 4 tokens in B columns.
// grid (IS/16), block = 16 waves; wave w reduces h in [w*128, w*128+128).
// ---------------------------------------------------------------------------
__global__ void shared_gateup_kernel(const float* __restrict__ x,
                                     const float* __restrict__ sgw,
                                     const float* __restrict__ suw,
                                     float* __restrict__ ws_y) {
  __shared__ float red_g[WAVES_SG * 32 * 8];   // 16KB
  __shared__ float red_u[WAVES_SG * 32 * 8];   // 16KB
  const int lane = threadIdx.x & 31;
  const int wv = threadIdx.x >> 5;
  const int L = lane & 15;         // A row (j offset) / B column (token)
  const int hi = lane >> 4;
  const int bk = hi * 2;
  const int jb = blockIdx.x * 16;

  const float* garow = sgw + (size_t)(jb + L) * H_DIM;
  const float* uarow = suw + (size_t)(jb + L) * H_DIM;
  const float* xrow = x + (size_t)L * H_DIM;

  const int h_beg = wv * (H_DIM / WAVES_SG);
  const int h_end = h_beg + (H_DIM / WAVES_SG);

  v8f dg = {};
  v8f du = {};
  for (int h0 = h_beg; h0 < h_end; h0 += 4) {
    v2f ag = *(const v2f*)(garow + h0 + bk);
    v2f au = *(const v2f*)(uarow + h0 + bk);
    v2f b;
    if (L < T_NUM) b = *(const v2f*)(xrow + h0 + bk);
    else { b.x = 0.0f; b.y = 0.0f; }
    dg = wmma_f32_16x16x4(ag, b, dg);
    du = wmma_f32_16x16x4(au, b, du);
  }
#pragma unroll
  for (int r = 0; r < 8; ++r) {
    red_g[(wv * 32 + lane) * 8 + r] = dg[r];
    red_u[(wv * 32 + lane) * 8 + r] = du[r];
  }
  __syncthreads();
  if (wv == 0) {
    v8f g = {};
    v8f u = {};
    for (int w = 0; w < WAVES_SG; ++w) {   // fixed order -> deterministic
#pragma unroll
      for (int r = 0; r < 8; ++r) {
        g[r] += red_g[(w * 32 + lane) * 8 + r];
        u[r] += red_u[(w * 32 + lane) * 8 + r];
      }
    }
    // D[m][n]: m = j offset (r + 8*hi), n = token (= L)
    if (L < T_NUM) {
      const int mofs = hi * 8;
#pragma unroll
      for (int r = 0; r < 8; ++r) {
        ws_y[(size_t)L * IS_DIM + jb + mofs + r] = silu_f(g[r]) * u[r];
      }
    }
  }
}

// ---------------------------------------------------------------------------
// Kernel 5: shared down ([H, IS], j contig -> weights in A) + combine with
// routed result; writes final output once.
// grid (H/16), block = 16 waves; wave w reduces j in [w*352, w*352+352).
// ---------------------------------------------------------------------------
__global__ void shared_down_combine_kernel(const float* __restrict__ sdw,
                                           const float* __restrict__ ws_y,
                                           const float* __restrict__ ws_routed,
                                           float* __restrict__ out) {
  __shared__ float red_d[WAVES_SD * 32 * 8];   // 16KB
  const int lane = threadIdx.x & 31;
  const int wv = threadIdx.x >> 5;
  const int L = lane & 15;
  const int hi = lane >> 4;
  const int bk = hi * 2;
  const int hb = blockIdx.x * 16;

  const float* arow = sdw + (size_t)(hb + L) * IS_DIM;
  const float* yrow = ws_y + (size_t)L * IS_DIM;

  const int j_beg = wv * (IS_DIM / WAVES_SD);
  const int j_end = j_beg + (IS_DIM / WAVES_SD);

  v8f d = {};
  for (int j0 = j_beg; j0 < j_end; j0 += 4) {
    v2f a = *(const v2f*)(arow + j0 + bk);
    v2f b;
    if (L < T_NUM) b = *(const v2f*)(yrow + j0 + bk);
    else { b.x = 0.0f; b.y = 0.0f; }
    d = wmma_f32_16x16x4(a, b, d);
  }
#pragma unroll
  for (int r = 0; r < 8; ++r) red_d[(wv * 32 + lane) * 8 + r] = d[r];
  __syncthreads();
  if (wv == 0) {
    v8f acc = {};
    for (int w = 0; w < WAVES_SD; ++w) {   // fixed order -> deterministic
#pragma unroll
      for (int r = 0; r < 8; ++r) acc[r] += red_d[(w * 32 + lane) * 8 + r];
    }
    // D[m][n]: m = h offset (r + 8*hi), n = token (= L)
    if (L < T_NUM) {
      const int mofs = hi * 8;
#pragma unroll
      for (int r = 0; r < 8; ++r) {
        const int h = hb + mofs + r;
        out[(size_t)L * H_DIM + h] = acc[r] + ws_routed[(size_t)L * H_DIM + h];
      }
    }
  }
}

// ---------------------------------------------------------------------------
extern "C" void kernel_launch(void* const* d_in, const int* in_sizes, int n_in,
                              void* d_out, int out_size, void* d_ws, size_t ws_size,
                              hipStream_t stream) {
  (void)in_sizes; (void)n_in; (void)out_size; (void)ws_size;
  const float* x   = (const float*)d_in[0];  // [4,1,2048]
  const float* rw  = (const float*)d_in[1];  // [16,2048]
  const float* gup = (const float*)d_in[2];  // [16,2048,2,1408]
  const float* dwn = (const float*)d_in[3];  // [16,1408,2048]
  const float* sgw = (const float*)d_in[4];  // [5632,2048]
  const float* suw = (const float*)d_in[5];  // [5632,2048]
  const float* sdw = (const float*)d_in[6];  // [2048,5632]
  float* out = (float*)d_out;

  // Workspace layout (all fully overwritten each call; no init needed):
  int*   ws_idx    = (int*)d_ws;                      // 8 ints
  float* wsf       = (float*)d_ws;
  float* ws_aff    = wsf + 8;                         // 8 floats
  float* ws_h      = wsf + 16;                        // 8 * 1408
  float* ws_y      = ws_h + 8 * I_DIM;                // 4 * 5632
  float* ws_routed = ws_y + 4 * IS_DIM;               // 4 * 2048

  router_topk_kernel<<<1, 64, 0, stream>>>(x, rw, ws_idx, ws_aff);
  routed_gateup_kernel<<<dim3(I_DIM / 16, T_NUM * TOPK), 32 * WAVES_RG, 0, stream>>>(
      x, gup, ws_idx, ws_aff, ws_h);
  routed_down_kernel<<<dim3(H_DIM / 16, T_NUM), 32 * WAVES_RD, 0, stream>>>(
      dwn, ws_idx, ws_h, ws_routed);
  shared_gateup_kernel<<<IS_DIM / 16, 32 * WAVES_SG, 0, stream>>>(x, sgw, suw, ws_y);
  shared_down_combine_kernel<<<H_DIM / 16, 32 * WAVES_SD, 0, stream>>>(
      sdw, ws_y, ws_routed, out);
}